// GNN_10393820857018
// MI455X (gfx1250) — compile-verified
//
#include <hip/hip_runtime.h>

typedef __attribute__((ext_vector_type(2))) float v2f;
typedef __attribute__((ext_vector_type(8))) float v8f;

#define WPB 2              // waves per block
#define TPB (WPB * 32)     // 64 threads, wave32

__device__ __forceinline__ void fatomic_add(float* p, float v) {
  unsafeAtomicAdd(p, v);   // -> global_atomic_add_f32
}

// ---------------------------------------------------------------------------
// Wave-level 16-row GEMM using V_WMMA_F32_16X16X4_F32 (fp32-exact path).
//   out[16][DOUT] = act(in[16][DIN] @ W[DIN][DOUT] + bias)
// Fragment layouts (wave32):
//   A v2f : lane l holds M = l&15,  K = k + 2*(l>>4) + {0,1}
//   B v2f : lane l holds N = n,     K = k + 2*(l>>4) + {0,1}
//   C/D v8f: lane l holds N = l&15, M = r + 8*(l>>4), r = 0..7
// EXEC is full for every wave that reaches this (uniform tile guard only).
// ---------------------------------------------------------------------------
template<int DIN, int DOUT, bool RELU>
__device__ __forceinline__ void wave_gemm16(const float* in,
                                            const float* __restrict__ W,
                                            const float* __restrict__ bias,
                                            float* out, int lane)
{
  const int half = lane >> 4;
  const int lm   = lane & 15;
#pragma unroll
  for (int nt = 0; nt < DOUT / 16; ++nt) {
    const int n = nt * 16 + lm;
    const float bv = bias[n];
    v8f acc = {bv, bv, bv, bv, bv, bv, bv, bv};
    for (int k = 0; k < DIN; k += 4) {
      const int kk = k + 2 * half;
      v2f a, b;
      a.x = in[lm * DIN + kk];
      a.y = in[lm * DIN + kk + 1];
      b.x = W[(size_t)kk * DOUT + n];
      b.y = W[(size_t)(kk + 1) * DOUT + n];
      acc = __builtin_amdgcn_wmma_f32_16x16x4_f32(false, a, false, b,
                                                  (short)0, acc, false, false);
    }
#pragma unroll
    for (int r = 0; r < 8; ++r) {
      float v = acc[r];
      if (RELU) v = fmaxf(v, 0.0f);
      out[(r + 8 * half) * DOUT + n] = v;
    }
  }
}

// Fused 2-layer MLP: io holds input [16][DIN] on entry, output [16][DOUT] on
// exit (io region must be 16*max(DIN,DOUT) floats). hid: 16*DH floats.
template<int DIN, int DH, int DOUT>
__device__ __forceinline__ void wave_mlp16(float* io, float* hid,
    const float* __restrict__ W1, const float* __restrict__ B1,
    const float* __restrict__ W2, const float* __restrict__ B2, int lane)
{
  wave_gemm16<DIN, DH, true >(io,  W1, B1, hid, lane);
  wave_gemm16<DH, DOUT, false>(hid, W2, B2, io,  lane);
}

// ---------------------------------------------------------------------------
// Utility kernels
// ---------------------------------------------------------------------------
__global__ void k_fill(float* __restrict__ p, long n, float v) {
  long i = (long)blockIdx.x * blockDim.x + threadIdx.x;
  long s = (long)gridDim.x * blockDim.x;
  for (; i < n; i += s) p[i] = v;
}

// in-degree per dst node, edges per batch (via btc[row]), nodes per batch
__global__ void k_counts(const int* __restrict__ row, const int* __restrict__ col,
                         const int* __restrict__ btc,
                         float* __restrict__ deg, float* __restrict__ ncnt,
                         float* __restrict__ ecnt, int E_, int N_)
{
  int i = blockIdx.x * blockDim.x + threadIdx.x;
  int s = gridDim.x * blockDim.x;
  for (int e = i; e < E_; e += s) {
    fatomic_add(&deg[col[e]], 1.0f);
    fatomic_add(&ecnt[btc[row[e]]], 1.0f);
  }
  for (int n = i; n < N_; n += s) fatomic_add(&ncnt[btc[n]], 1.0f);
}

// ---------------------------------------------------------------------------
// Dense row-wise MLP (encoders): dst[r] = mlp(src[r]); src==dst allowed.
// ---------------------------------------------------------------------------
template<int DIN, int DH, int DOUT>
__global__ void __launch_bounds__(TPB)
k_dense_mlp(const float* src, float* dst, int nrows,
            const float* __restrict__ W1, const float* __restrict__ B1,
            const float* __restrict__ W2, const float* __restrict__ B2)
{
  constexpr int IOW = (DIN > DOUT) ? DIN : DOUT;
  __shared__ float lds[WPB * 16 * (IOW + DH)];
  const int lane = threadIdx.x & 31, wid = threadIdx.x >> 5;
  const int tile = blockIdx.x * WPB + wid;
  if (tile * 16 >= nrows) return;
  float* io  = lds + wid * 16 * (IOW + DH);
  float* hid = io + 16 * IOW;
  const int rb = tile * 16;
  for (int r = 0; r < 16; ++r)
    for (int j = lane; j < DIN; j += 32)
      io[r * DIN + j] = src[(size_t)(rb + r) * DIN + j];
  wave_mlp16<DIN, DH, DOUT>(io, hid, W1, B1, W2, B2, lane);
  for (int r = 0; r < 16; ++r)
    for (int j = lane; j < DOUT; j += 32)
      dst[(size_t)(rb + r) * DOUT + j] = io[r * DOUT + j];
}

// ---------------------------------------------------------------------------
// Stage-1 meta kernels
// ---------------------------------------------------------------------------
// edge update: e[ei] = mlp(edge1, [x[row], x[col], e[ei]]) (in place), fused
// scatter-add into agg[col] (for node seg-mean) and esum[btc[row]] (globals)
__global__ void __launch_bounds__(TPB)
k_edge1(const float* __restrict__ x, float* ef,
        const int* __restrict__ row, const int* __restrict__ col,
        const int* __restrict__ btc,
        const float* __restrict__ W1, const float* __restrict__ B1,
        const float* __restrict__ W2, const float* __restrict__ B2,
        float* __restrict__ agg, float* __restrict__ esum, int nedges)
{
  constexpr int DIN = 320, DH = 64;
  __shared__ float lds[WPB * 16 * (DIN + DH)];
  const int lane = threadIdx.x & 31, wid = threadIdx.x >> 5;
  const int tile = blockIdx.x * WPB + wid;
  if (tile * 16 >= nedges) return;
  float* io  = lds + wid * 16 * (DIN + DH);
  float* hid = io + 16 * DIN;
  const int eb = tile * 16;
  for (int e = 0; e < 16; ++e) {
    const int ei = eb + e;
    const int r = row[ei], c = col[ei];
    for (int j = lane; j < 128; j += 32) io[e * DIN + j]       = x[(size_t)r * 128 + j];
    for (int j = lane; j < 128; j += 32) io[e * DIN + 128 + j] = x[(size_t)c * 128 + j];
    for (int j = lane; j < 64;  j += 32) io[e * DIN + 256 + j] = ef[(size_t)ei * 64 + j];
  }
  wave_mlp16<DIN, DH, 64>(io, hid, W1, B1, W2, B2, lane);
  for (int idx = lane; idx < 16 * 64; idx += 32) {
    const int e = idx >> 6, f = idx & 63;
    const int ei = eb + e;
    const float v = io[e * 64 + f];
    ef[(size_t)ei * 64 + f] = v;
    fatomic_add(&agg[(size_t)col[ei] * 64 + f], v);
    fatomic_add(&esum[(size_t)btc[row[ei]] * 64 + f], v);
  }
}

// node update: x[n] = mlp(node1, [x[n], agg[n]/max(deg,1)]) in place,
// fused scatter-add of x2 into xsum[btc[n]]
__global__ void __launch_bounds__(TPB)
k_node1(float* x, const float* __restrict__ agg, const float* __restrict__ deg,
        const int* __restrict__ btc,
        const float* __restrict__ W1, const float* __restrict__ B1,
        const float* __restrict__ W2, const float* __restrict__ B2,
        float* __restrict__ xsum, int nnodes)
{
  constexpr int DIN = 192, DH = 128;
  __shared__ float lds[WPB * 16 * (DIN + DH)];
  const int lane = threadIdx.x & 31, wid = threadIdx.x >> 5;
  const int tile = blockIdx.x * WPB + wid;
  if (tile * 16 >= nnodes) return;
  float* io  = lds + wid * 16 * (DIN + DH);
  float* hid = io + 16 * DIN;
  const int nb = tile * 16;
  for (int r = 0; r < 16; ++r) {
    const int n = nb + r;
    const float inv = 1.0f / fmaxf(deg[n], 1.0f);
    for (int j = lane; j < 128; j += 32) io[r * DIN + j]       = x[(size_t)n * 128 + j];
    for (int j = lane; j < 64;  j += 32) io[r * DIN + 128 + j] = agg[(size_t)n * 64 + j] * inv;
  }
  wave_mlp16<DIN, DH, 128>(io, hid, W1, B1, W2, B2, lane);
  for (int idx = lane; idx < 16 * 128; idx += 32) {
    const int r = idx >> 7, f = idx & 127;
    const int n = nb + r;
    const float v = io[r * 128 + f];
    x[(size_t)n * 128 + f] = v;
    fatomic_add(&xsum[(size_t)btc[n] * 128 + f], v);
  }
}

// global update: u[b] = mlp(glob1, [u, xsum/ncnt, esum/ecnt]) in place
__global__ void __launch_bounds__(TPB)
k_glob1(float* u, const float* __restrict__ xsum, const float* __restrict__ esum,
        const float* __restrict__ ncnt, const float* __restrict__ ecnt,
        const float* __restrict__ W1, const float* __restrict__ B1,
        const float* __restrict__ W2, const float* __restrict__ B2, int nrows)
{
  constexpr int DIN = 256, DH = 64;
  __shared__ float lds[WPB * 16 * (DIN + DH)];
  const int lane = threadIdx.x & 31, wid = threadIdx.x >> 5;
  const int tile = blockIdx.x * WPB + wid;
  if (tile * 16 >= nrows) return;
  float* io  = lds + wid * 16 * (DIN + DH);
  float* hid = io + 16 * DIN;
  const int bb = tile * 16;
  for (int r = 0; r < 16; ++r) {
    const int b = bb + r;
    const float invn = 1.0f / fmaxf(ncnt[b], 1.0f);
    const float inve = 1.0f / fmaxf(ecnt[b], 1.0f);
    for (int j = lane; j < 64;  j += 32) io[r * DIN + j]       = u[(size_t)b * 64 + j];
    for (int j = lane; j < 128; j += 32) io[r * DIN + 64 + j]  = xsum[(size_t)b * 128 + j] * invn;
    for (int j = lane; j < 64;  j += 32) io[r * DIN + 192 + j] = esum[(size_t)b * 64 + j] * inve;
  }
  wave_mlp16<DIN, DH, 64>(io, hid, W1, B1, W2, B2, lane);
  for (int r = 0; r < 16; ++r)
    for (int j = lane; j < 64; j += 32)
      u[(size_t)(bb + r) * 64 + j] = io[r * 64 + j];
}

// ---------------------------------------------------------------------------
// Stage-2 setup: c1/c2 = mlp(enc_e2, [T, u_px/u_pm]) (K=2: scalar path),
// build eg[4B,32] = {c1,c1,c2,c2}, xg[3B,64] = {u1,u2,um}, ug = 0.1
// ---------------------------------------------------------------------------
__global__ void k_stage2_init(const float* __restrict__ temp,
                              const float* __restrict__ up1, const float* __restrict__ up2,
                              const float* __restrict__ upm, const float* __restrict__ uall,
                              const float* __restrict__ W1, const float* __restrict__ B1,
                              const float* __restrict__ W2, const float* __restrict__ B2,
                              float* __restrict__ xg, float* __restrict__ eg,
                              float* __restrict__ ug, int B_)
{
  const int b = blockIdx.x * blockDim.x + threadIdx.x;
  if (b >= B_) return;
  const float t = temp[b];
  const float invm = 1.0f / upm[b];
#pragma unroll
  for (int p = 0; p < 2; ++p) {
    const float rr = (p ? up2[b] : up1[b]) * invm;
    float h[32];
#pragma unroll
    for (int j = 0; j < 32; ++j)
      h[j] = fmaxf(t * W1[j] + rr * W1[32 + j] + B1[j], 0.0f);
#pragma unroll
    for (int o = 0; o < 32; ++o) {
      float s = B2[o];
#pragma unroll
      for (int j = 0; j < 32; ++j) s += h[j] * W2[j * 32 + o];
      eg[(size_t)(b * 4 + 2 * p) * 32 + o]     = s;
      eg[(size_t)(b * 4 + 2 * p + 1) * 32 + o] = s;
    }
  }
  for (int k = 0; k < 3; ++k)
    for (int f = 0; f < 64; ++f)
      xg[(size_t)(3 * b + k) * 64 + f] = uall[(size_t)k * B_ * 64 + (size_t)b * 64 + f];
  for (int f = 0; f < 64; ++f) ug[(size_t)b * 64 + f] = 0.1f;
}

// stage-2 edge update (connectivity is fixed: row={0,2,1,2}, col={2,0,2,1} +3b)
__global__ void __launch_bounds__(TPB)
k_edge2(const float* __restrict__ xg, float* eg,
        const float* __restrict__ W1, const float* __restrict__ B1,
        const float* __restrict__ W2, const float* __restrict__ B2,
        float* __restrict__ agg2, float* __restrict__ esum2, int nedges)
{
  constexpr int DIN = 160, DH = 32;
  __shared__ float lds[WPB * 16 * (DIN + DH)];
  const int lane = threadIdx.x & 31, wid = threadIdx.x >> 5;
  const int tile = blockIdx.x * WPB + wid;
  if (tile * 16 >= nedges) return;
  float* io  = lds + wid * 16 * (DIN + DH);
  float* hid = io + 16 * DIN;
  const int eb = tile * 16;
  for (int e = 0; e < 16; ++e) {
    const int ei = eb + e, b = ei >> 2, j = ei & 3;
    const int r = 3 * b + ((152 >> (2 * j)) & 3);   // packed {0,2,1,2}
    const int c = 3 * b + ((98  >> (2 * j)) & 3);   // packed {2,0,2,1}
    for (int f = lane; f < 64; f += 32) io[e * DIN + f]       = xg[(size_t)r * 64 + f];
    for (int f = lane; f < 64; f += 32) io[e * DIN + 64 + f]  = xg[(size_t)c * 64 + f];
    for (int f = lane; f < 32; f += 32) io[e * DIN + 128 + f] = eg[(size_t)ei * 32 + f];
  }
  wave_mlp16<DIN, DH, 32>(io, hid, W1, B1, W2, B2, lane);
  for (int idx = lane; idx < 16 * 32; idx += 32) {
    const int e = idx >> 5, f = idx & 31;
    const int ei = eb + e, b = ei >> 2, j = ei & 3;
    const int c = 3 * b + ((98 >> (2 * j)) & 3);
    const float v = io[e * 32 + f];
    eg[(size_t)ei * 32 + f] = v;
    fatomic_add(&agg2[(size_t)c * 32 + f], v);
    fatomic_add(&esum2[(size_t)b * 32 + f], v);
  }
}

// stage-2 node update (in-degree: local node 2 has 2 incoming, others 1)
__global__ void __launch_bounds__(TPB)
k_node2(float* xg, const float* __restrict__ agg2,
        const float* __restrict__ W1, const float* __restrict__ B1,
        const float* __restrict__ W2, const float* __restrict__ B2,
        float* __restrict__ xsum2, int nnodes)
{
  constexpr int DIN = 96, DH = 64;
  __shared__ float lds[WPB * 16 * (DIN + DH)];
  const int lane = threadIdx.x & 31, wid = threadIdx.x >> 5;
  const int tile = blockIdx.x * WPB + wid;
  if (tile * 16 >= nnodes) return;
  float* io  = lds + wid * 16 * (DIN + DH);
  float* hid = io + 16 * DIN;
  const int nb = tile * 16;
  for (int r = 0; r < 16; ++r) {
    const int n = nb + r, b = n / 3, l = n - 3 * b;
    const float inv = (l == 2) ? 0.5f : 1.0f;
    for (int f = lane; f < 64; f += 32) io[r * DIN + f]      = xg[(size_t)n * 64 + f];
    for (int f = lane; f < 32; f += 32) io[r * DIN + 64 + f] = agg2[(size_t)n * 32 + f] * inv;
  }
  wave_mlp16<DIN, DH, 64>(io, hid, W1, B1, W2, B2, lane);
  for (int idx = lane; idx < 16 * 64; idx += 32) {
    const int r = idx >> 6, f = idx & 63;
    const int n = nb + r, b = n / 3;
    const float v = io[r * 64 + f];
    xg[(size_t)n * 64 + f] = v;
    fatomic_add(&xsum2[(size_t)b * 64 + f], v);
  }
}

// stage-2 global update: exactly 3 nodes / 4 edges per batch element
__global__ void __launch_bounds__(TPB)
k_glob2(float* ug, const float* __restrict__ xsum2, const float* __restrict__ esum2,
        const float* __restrict__ W1, const float* __restrict__ B1,
        const float* __restrict__ W2, const float* __restrict__ B2, int nrows)
{
  constexpr int DIN = 160, DH = 64;
  __shared__ float lds[WPB * 16 * (DIN + DH)];
  const int lane = threadIdx.x & 31, wid = threadIdx.x >> 5;
  const int tile = blockIdx.x * WPB + wid;
  if (tile * 16 >= nrows) return;
  float* io  = lds + wid * 16 * (DIN + DH);
  float* hid = io + 16 * DIN;
  const int bb = tile * 16;
  for (int r = 0; r < 16; ++r) {
    const int b = bb + r;
    for (int f = lane; f < 64; f += 32) io[r * DIN + f]       = ug[(size_t)b * 64 + f];
    for (int f = lane; f < 64; f += 32) io[r * DIN + 64 + f]  = xsum2[(size_t)b * 64 + f] * (1.0f / 3.0f);
    for (int f = lane; f < 32; f += 32) io[r * DIN + 128 + f] = esum2[(size_t)b * 32 + f] * 0.25f;
  }
  wave_mlp16<DIN, DH, 64>(io, hid, W1, B1, W2, B2, lane);
  for (int r = 0; r < 16; ++r)
    for (int f = lane; f < 64; f += 32)
      ug[(size_t)(bb + r) * 64 + f] = io[r * 64 + f];
}

// final head: 64 -> 64 (relu) -> 15; second layer runs as a 16-wide WMMA tile
// with column 15 masked (zero weights/bias, store suppressed).
__global__ void __launch_bounds__(TPB)
k_last(const float* __restrict__ ug, float* __restrict__ out,
       const float* __restrict__ W1, const float* __restrict__ B1,
       const float* __restrict__ W2, const float* __restrict__ B2, int nrows)
{
  __shared__ float lds[WPB * 16 * (64 + 64)];
  const int lane = threadIdx.x & 31, wid = threadIdx.x >> 5;
  const int tile = blockIdx.x * WPB + wid;
  if (tile * 16 >= nrows) return;
  float* io  = lds + wid * 16 * 128;
  float* hid = io + 16 * 64;
  const int rb = tile * 16;
  for (int r = 0; r < 16; ++r)
    for (int j = lane; j < 64; j += 32) io[r * 64 + j] = ug[(size_t)(rb + r) * 64 + j];
  wave_gemm16<64, 64, true>(io, W1, B1, hid, lane);
  const int half = lane >> 4, lm = lane & 15;
  const bool valid = (lm < 15);
  const float bv = valid ? B2[lm] : 0.0f;
  v8f acc = {bv, bv, bv, bv, bv, bv, bv, bv};
  for (int k = 0; k < 64; k += 4) {
    const int kk = k + 2 * half;
    v2f a, b;
    a.x = hid[lm * 64 + kk];
    a.y = hid[lm * 64 + kk + 1];
    b.x = valid ? W2[kk * 15 + lm] : 0.0f;
    b.y = valid ? W2[(kk + 1) * 15 + lm] : 0.0f;
    acc = __builtin_amdgcn_wmma_f32_16x16x4_f32(false, a, false, b,
                                                (short)0, acc, false, false);
  }
#pragma unroll
  for (int r = 0; r < 8; ++r)
    if (valid) out[(size_t)(rb + r + 8 * half) * 15 + lm] = acc[r];
}

// ---------------------------------------------------------------------------
// Host orchestration
// ---------------------------------------------------------------------------
static inline int tile_blocks(int rows) {
  int t = (rows + 15) / 16;
  return (t + WPB - 1) / WPB;
}

#define MLP4(name, i)                                   \
  const float* name##_W1 = (const float*)d_in[(i)];     \
  const float* name##_B1 = (const float*)d_in[(i) + 1]; \
  const float* name##_W2 = (const float*)d_in[(i) + 2]; \
  const float* name##_B2 = (const float*)d_in[(i) + 3];

extern "C" void kernel_launch(void* const* d_in, const int* in_sizes, int n_in,
                              void* d_out, int out_size, void* d_ws, size_t ws_size,
                              hipStream_t stream)
{
  const int N_ = in_sizes[0] / 32;   // nodes
  const int E_ = in_sizes[3] / 16;   // edges
  const int Bb = in_sizes[6];        // batch

  MLP4(encn1, 16) MLP4(ence1, 20) MLP4(encn2, 24) MLP4(ence2, 28)
  MLP4(edge1, 32) MLP4(node1, 36) MLP4(glob1, 40)
  MLP4(edge2, 44) MLP4(node2, 48) MLP4(glob2, 52) MLP4(lastm, 56)

  const float* up1  = (const float*)d_in[6];
  const float* up2  = (const float*)d_in[7];
  const float* upm  = (const float*)d_in[8];
  const float* temp = (const float*)d_in[9];

  // --- workspace layout (floats) ---
  float* ws = (float*)d_ws;
  size_t off = 0;
  auto walloc = [&](size_t n) { float* p = ws + off; off += (n + 63) & ~(size_t)63; return p; };
  float* X     = walloc((size_t)N_ * 128);
  float* EF    = walloc((size_t)E_ * 64);
  float* AGG   = walloc((size_t)N_ * 64);
  float* DEG   = walloc(N_);
  float* NCNT  = walloc(Bb);
  float* ECNT  = walloc(Bb);
  float* XSUM  = walloc((size_t)Bb * 128);
  float* ESUM  = walloc((size_t)Bb * 64);
  float* UALL  = walloc((size_t)3 * Bb * 64);
  float* XG    = walloc((size_t)3 * Bb * 64);
  float* EG    = walloc((size_t)4 * Bb * 32);
  float* UG    = walloc((size_t)Bb * 64);
  float* AGG2  = walloc((size_t)3 * Bb * 32);
  float* XSUM2 = walloc((size_t)Bb * 64);
  float* ESUM2 = walloc((size_t)Bb * 32);
  if (off * sizeof(float) > ws_size) return;  // insufficient scratch

  auto fill = [&](float* p, long n, float v) {
    int blk = (int)((n + 255) / 256);
    if (blk > 2048) blk = 2048;
    k_fill<<<blk, 256, 0, stream>>>(p, n, v);
  };

  const int nB = tile_blocks(N_), eB = tile_blocks(E_), bB = tile_blocks(Bb);

  // ---- stage 1: three graphs processed sequentially (reuse X/EF/AGG) ----
  for (int g = 0; g < 3; ++g) {
    const float* xin = (const float*)d_in[0 + g];
    const float* ein = (const float*)d_in[3 + g];
    const int*   ei  = (const int*)d_in[10 + g];
    const int*   row = ei;
    const int*   col = ei + E_;
    const int*   btc = (const int*)d_in[13 + g];
    float* Ug = UALL + (size_t)g * Bb * 64;

    fill(Ug, (long)Bb * 64, 0.1f);
    fill(DEG, N_, 0.0f); fill(NCNT, Bb, 0.0f); fill(ECNT, Bb, 0.0f);
    k_counts<<<1024, 256, 0, stream>>>(row, col, btc, DEG, NCNT, ECNT, E_, N_);

    k_dense_mlp<32, 128, 128><<<nB, TPB, 0, stream>>>(xin, X, N_,
        encn1_W1, encn1_B1, encn1_W2, encn1_B2);
    k_dense_mlp<16, 64, 64><<<eB, TPB, 0, stream>>>(ein, EF, E_,
        ence1_W1, ence1_B1, ence1_W2, ence1_B2);

    for (int round = 0; round < 2; ++round) {
      fill(AGG, (long)N_ * 64, 0.0f);
      fill(ESUM, (long)Bb * 64, 0.0f);
      fill(XSUM, (long)Bb * 128, 0.0f);
      k_edge1<<<eB, TPB, 0, stream>>>(X, EF, row, col, btc,
          edge1_W1, edge1_B1, edge1_W2, edge1_B2, AGG, ESUM, E_);
      k_node1<<<nB, TPB, 0, stream>>>(X, AGG, DEG, btc,
          node1_W1, node1_B1, node1_W2, node1_B2, XSUM, N_);
      k_glob1<<<bB, TPB, 0, stream>>>(Ug, XSUM, ESUM, NCNT, ECNT,
          glob1_W1, glob1_B1, glob1_W2, glob1_B2, Bb);
    }
    k_dense_mlp<64, 64, 64><<<bB, TPB, 0, stream>>>(Ug, Ug, Bb,
        encn2_W1, encn2_B1, encn2_W2, encn2_B2);
  }

  // ---- stage 2: tiny 3-node/4-edge graphs per batch element ----
  k_stage2_init<<<(Bb + 63) / 64, 64, 0, stream>>>(temp, up1, up2, upm, UALL,
      ence2_W1, ence2_B1, ence2_W2, ence2_B2, XG, EG, UG, Bb);

  const int e2B = tile_blocks(4 * Bb), n2B = tile_blocks(3 * Bb);
  for (int round = 0; round < 2; ++round) {
    fill(AGG2, (long)3 * Bb * 32, 0.0f);
    fill(XSUM2, (long)Bb * 64, 0.0f);
    fill(ESUM2, (long)Bb * 32, 0.0f);
    k_edge2<<<e2B, TPB, 0, stream>>>(XG, EG,
        edge2_W1, edge2_B1, edge2_W2, edge2_B2, AGG2, ESUM2, 4 * Bb);
    k_node2<<<n2B, TPB, 0, stream>>>(XG, AGG2,
        node2_W1, node2_B1, node2_W2, node2_B2, XSUM2, 3 * Bb);
    k_glob2<<<bB, TPB, 0, stream>>>(UG, XSUM2, ESUM2,
        glob2_W1, glob2_B1, glob2_W2, glob2_B2, Bb);
  }

  k_last<<<bB, TPB, 0, stream>>>(UG, (float*)d_out,
      lastm_W1, lastm_B1, lastm_W2, lastm_B2, Bb);
}